// PermopRagged_16552803958995
// MI455X (gfx1250) — compile-verified
//
#include <hip/hip_runtime.h>
#include <stdint.h>

// Problem constants (from reference): values (P=1048576, C=256) f32,
// sorted segment ids, B=512 segments, out (512, 256) f32.
constexpr int CDIM  = 256;   // channels per row
constexpr int HALF  = 128;   // floats per half-row (one wave's slice)
constexpr int WAVES = 8;     // waves per block (256 threads, wave32)
constexpr int BATCH = 8;     // rows per async batch
constexpr int ROWS_PER_WAVE = 512;

__global__ void zero_out_kernel(float* __restrict__ out, int n) {
    int i = blockIdx.x * blockDim.x + threadIdx.x;
    if (i < n) out[i] = 0.0f;
}

// One async DMA of a 512B row-half into LDS: each of the 32 lanes moves 16B.
// Non-temporal: the 1 GiB values stream is touched exactly once (5x L2 size),
// so keep it from evicting seg-ids / output lines that do have reuse.
__device__ __forceinline__ void async_row_load(unsigned lds_off, unsigned voff,
                                               const float* sbase) {
    asm volatile("global_load_async_to_lds_b128 %0, %1, %2 th:TH_LOAD_NT"
                 :
                 : "v"(lds_off), "v"(voff), "s"(sbase)
                 : "memory");
}

__global__ void __launch_bounds__(WAVES * 32)
segsum_kernel(const float* __restrict__ vals,
              const int*   __restrict__ segs,
              float* __restrict__ out,
              int total_rows) {
    // 8 waves * 2 buffers * 8 rows * 512B = 64 KB
    __shared__ float smem[WAVES][2][BATCH][HALF];

    const int wave = threadIdx.x >> 5;
    const int lane = threadIdx.x & 31;

    // Wave-task: (row chunk, channel half)
    const int  task  = blockIdx.x * WAVES + wave;
    const int  half  = task & 1;
    const int  chunk = task >> 1;
    const long row0  = (long)chunk * ROWS_PER_WAVE;
    if (row0 >= total_rows) return;

    const float*   gbase  = vals + row0 * CDIM + half * HALF; // wave-uniform
    const unsigned lane_b = (unsigned)lane * 16u;             // byte offset in 512B slot

    auto issue_batch = [&](int bi, int parity) {
        const unsigned r0 = (unsigned)bi * BATCH;
#pragma unroll
        for (int s = 0; s < BATCH; ++s) {
            unsigned lds_off =
                (unsigned)(uintptr_t)(&smem[wave][parity][s][0]) + lane_b;
            unsigned voff = (r0 + (unsigned)s) * (unsigned)(CDIM * 4) + lane_b;
            async_row_load(lds_off, voff, gbase);
        }
    };

    int    cur = segs[row0];              // wave-uniform (sorted ids)
    float4 acc = make_float4(0.f, 0.f, 0.f, 0.f);

    issue_batch(0, 0);
    issue_batch(1, 1);

    constexpr int NB = ROWS_PER_WAVE / BATCH;
    for (int b = 0; b < NB; ++b) {
        // Oldest batch complete: outstanding drops to <= BATCH (or 0 at drain).
        if (b + 1 < NB)
            asm volatile("s_wait_asynccnt 0x8" ::: "memory");
        else
            asm volatile("s_wait_asynccnt 0x0" ::: "memory");

        const int parity = b & 1;
#pragma unroll
        for (int s = 0; s < BATCH; ++s) {
            const long r = row0 + (long)b * BATCH + s;
            const float4 v =
                *reinterpret_cast<const float4*>(&smem[wave][parity][s][lane * 4]);
            const int sg = segs[r];
            if (sg != cur) {  // rare: segment boundary -> flush accumulator
                float* o = out + (size_t)cur * CDIM + half * HALF + lane * 4;
                unsafeAtomicAdd(o + 0, acc.x);
                unsafeAtomicAdd(o + 1, acc.y);
                unsafeAtomicAdd(o + 2, acc.z);
                unsafeAtomicAdd(o + 3, acc.w);
                acc = make_float4(0.f, 0.f, 0.f, 0.f);
                cur = sg;
            }
            acc.x += v.x; acc.y += v.y; acc.z += v.z; acc.w += v.w;
        }

        // LDS reads (DScnt) are unordered vs async LDS writes (ASYNCcnt):
        // drain ds reads before overwriting this buffer.
        asm volatile("s_wait_dscnt 0x0" ::: "memory");
        if (b + 2 < NB) issue_batch(b + 2, parity);
    }

    // Final flush
    {
        float* o = out + (size_t)cur * CDIM + half * HALF + lane * 4;
        unsafeAtomicAdd(o + 0, acc.x);
        unsafeAtomicAdd(o + 1, acc.y);
        unsafeAtomicAdd(o + 2, acc.z);
        unsafeAtomicAdd(o + 3, acc.w);
    }
}

extern "C" void kernel_launch(void* const* d_in, const int* in_sizes, int n_in,
                              void* d_out, int out_size, void* d_ws, size_t ws_size,
                              hipStream_t stream) {
    const float* vals = (const float*)d_in[0];
    const int*   segs = (const int*)d_in[1];
    float*       out  = (float*)d_out;

    const int total_rows = in_sizes[0] / CDIM;   // 1048576

    // Zero the output (harness poisons d_out; atomics accumulate on top).
    {
        int threads = 256;
        int blocks  = (out_size + threads - 1) / threads;
        zero_out_kernel<<<blocks, threads, 0, stream>>>(out, out_size);
    }

    // One wave-task per (512-row chunk, channel half).
    const int chunks = (total_rows + ROWS_PER_WAVE - 1) / ROWS_PER_WAVE;
    const int tasks  = chunks * 2;
    const int blocks = (tasks + WAVES - 1) / WAVES;   // 512 blocks
    segsum_kernel<<<blocks, WAVES * 32, 0, stream>>>(vals, segs, out, total_rows);
}